// GenericGAT_33440615367356
// MI455X (gfx1250) — compile-verified
//
#include <hip/hip_runtime.h>
#include <hip/hip_bf16.h>
#include <math.h>

typedef __attribute__((ext_vector_type(2))) float v2f;
typedef __attribute__((ext_vector_type(8))) float v8f;

#define NEG_SLOPE 0.2f

// ---------------------------------------------------------------------------
// float atomic max via ordered-int trick (buffers initialized to -INF)
// ---------------------------------------------------------------------------
__device__ __forceinline__ void atomicMaxF32(float* addr, float val) {
  if (val >= 0.0f) {
    atomicMax((int*)addr, __float_as_int(val));
  } else {
    atomicMin((unsigned int*)addr, (unsigned int)__float_as_int(val));
  }
}

// ---------------------------------------------------------------------------
// fill
// ---------------------------------------------------------------------------
__global__ void __launch_bounds__(256)
fill_kernel(float* __restrict__ p, float v, int n) {
  int i = blockIdx.x * 256 + threadIdx.x;
  if (i < n) p[i] = v;
}

// ---------------------------------------------------------------------------
// fp32 GEMM via V_WMMA_F32_16X16X4_F32.
// Block = 8 waves = one 128-row M-strip x one 16-col N-tile.
// The K x 16 B panel is staged ONCE into LDS with GLOBAL_LOAD_ASYNC_TO_LDS_B32
// (per-lane LDS dest addresses do the pair swizzle during the copy), packed so
// that the two B values a lane needs per WMMA step are adjacent:
//   lds[(k/2)*32 + n*2 + (k&1)] = B[k][tn*16+n]
// Inner loop per WMMA: 1x global b64 (A pair) + 1x ds b64 (B pair).
//
// WMMA f32 16x16x4 operand layouts (ISA 7.12.2):
//   A: lane L<16 -> A[M=L, K={k0,k0+1}]; lane L>=16 -> A[M=L-16, K={k0+2,k0+3}]
//   B: VGPR0 lanes0-15 = row k0, lanes16-31 = row k0+2; VGPR1 = k0+1 / k0+3
//   C: VGPR r: lanes0-15 -> M=r, lanes16-31 -> M=r+8; N = lane&15
// ---------------------------------------------------------------------------
__global__ void __launch_bounds__(256)
gemm_wmma_f32(const float* __restrict__ A, const float* __restrict__ B,
              float* __restrict__ C, int M, int K, int Nout) {
  __shared__ float bpan[256 * 16];              // max K = 256 -> 16 KB

  const int tilesM = M >> 4;
  const int tilesN = Nout >> 4;
  const int strip  = blockIdx.x / tilesN;       // 8 M-tiles per strip
  const int tn     = blockIdx.x - strip * tilesN;
  const int wave   = threadIdx.x >> 5;
  const int lane   = threadIdx.x & 31;

  // ---- stage B panel (K x 16) into LDS via async copy, pair-swizzled ----
  {
    const int colbase = tn * 16;
    for (int idx = threadIdx.x; idx < K * 16; idx += 256) {
      const int k = idx >> 4;
      const int n = idx & 15;
      const float* gp = B + (size_t)k * Nout + colbase + n;
      unsigned int ldsAddr =
          (unsigned int)(uintptr_t)(&bpan[((k >> 1) * 32) + n * 2 + (k & 1)]);
      asm volatile("global_load_async_to_lds_b32 %0, %1, off"
                   :: "v"(ldsAddr), "v"(gp)
                   : "memory");
    }
    asm volatile("s_wait_asynccnt 0x0" ::: "memory");
  }
  __syncthreads();

  // ---- compute (wave-uniform guard: EXEC all-ones around WMMA) ----
  const int tm = strip * 8 + wave;
  if (tm < tilesM) {
    const int row   = tm * 16 + (lane & 15);
    const int col   = tn * 16 + (lane & 15);
    const int khalf = (lane >= 16) ? 2 : 0;
    const int nn    = (lane & 15) * 2;

    const float* Arow = A + (size_t)row * K + khalf;
    v8f acc = {};
    for (int k0 = 0; k0 < K; k0 += 4) {
      v2f a = *(const v2f*)(Arow + k0);                         // 8B aligned
      const int kk = k0 + khalf;                                // even
      v2f b = *(const v2f*)(&bpan[(kk >> 1) * 32 + nn]);        // ds_load_b64
      acc = __builtin_amdgcn_wmma_f32_16x16x4_f32(
          /*neg_a=*/false, a, /*neg_b=*/false, b,
          /*c_mod=*/(short)0, acc, /*reuse_a=*/false, /*reuse_b=*/false);
    }

    const int rbase = tm * 16 + ((lane >= 16) ? 8 : 0);
    float* Cp = C + (size_t)rbase * Nout + col;
#pragma unroll
    for (int r = 0; r < 8; ++r) Cp[(size_t)r * Nout] = acc[r];
  }
}

// ---------------------------------------------------------------------------
// per-node attention logits: as[n,h] = dot(H[n,h,:], a_src[h,:]); same for ad
// ---------------------------------------------------------------------------
__global__ void __launch_bounds__(256)
alpha_kernel(const float* __restrict__ Hm, const float* __restrict__ a_s,
             const float* __restrict__ a_d, float* __restrict__ as_o,
             float* __restrict__ ad_o, int N, int Hh, int D) {
  int idx = blockIdx.x * 256 + threadIdx.x;
  if (idx >= N * Hh) return;
  int n = idx / Hh, h = idx - n * Hh;
  const float* hp  = Hm + (size_t)n * Hh * D + (size_t)h * D;
  const float* asp = a_s + (size_t)h * D;
  const float* adp = a_d + (size_t)h * D;
  float s = 0.0f, d = 0.0f;
  for (int i = 0; i < D; ++i) {
    float v = hp[i];
    s += v * asp[i];
    d += v * adp[i];
  }
  as_o[idx] = s;
  ad_o[idx] = d;
}

// ---------------------------------------------------------------------------
// edge pass 1: e = leaky_relu(as[src]+ad[dst]); segment max over dst
// edges e < E come from edge_index; e >= E are self-loops (src=dst=e-E)
// ---------------------------------------------------------------------------
__global__ void __launch_bounds__(256)
edge_logits_max(const int* __restrict__ ei, int E, int N, int Hh,
                const float* __restrict__ as_, const float* __restrict__ ad_,
                float* __restrict__ ebuf, float* __restrict__ maxb) {
  int ET = E + N;
  int idx = blockIdx.x * 256 + threadIdx.x;
  if (idx >= ET * Hh) return;
  int e = idx / Hh, h = idx - e * Hh;
  int s, d;
  if (e < E) { s = ei[e]; d = ei[E + e]; } else { s = e - E; d = s; }
  float v = as_[s * Hh + h] + ad_[d * Hh + h];
  v = (v > 0.0f) ? v : NEG_SLOPE * v;
  ebuf[idx] = v;
  atomicMaxF32(&maxb[d * Hh + h], v);
}

// ---------------------------------------------------------------------------
// edge pass 2: ex = exp(e - max[dst]); segment sum of ex over dst
// ---------------------------------------------------------------------------
__global__ void __launch_bounds__(256)
edge_exp_sum(const int* __restrict__ ei, int E, int N, int Hh,
             float* __restrict__ ebuf, const float* __restrict__ maxb,
             float* __restrict__ den) {
  int ET = E + N;
  int idx = blockIdx.x * 256 + threadIdx.x;
  if (idx >= ET * Hh) return;
  int e = idx / Hh, h = idx - e * Hh;
  int d = (e < E) ? ei[E + e] : (e - E);
  float ex = __expf(ebuf[idx] - maxb[d * Hh + h]);
  ebuf[idx] = ex;
  atomicAdd(&den[d * Hh + h], ex);
}

// ---------------------------------------------------------------------------
// edge aggregation: acc[dst, f] += H[src, f] * ex[e, f/D]
// F = Hh*D features per node; 256/F edges per 256-thread block -> gathers and
// atomic adds are contiguous per edge row.
// ---------------------------------------------------------------------------
__global__ void __launch_bounds__(256)
edge_aggregate(const int* __restrict__ ei, int E, int N, int Hh, int D,
               const float* __restrict__ Hm, const float* __restrict__ ebuf,
               float* __restrict__ acc) {
  const int F = Hh * D;
  const int epb = 256 / F;
  const int le = threadIdx.x / F;
  const int f  = threadIdx.x - le * F;
  const int e  = blockIdx.x * epb + le;
  const int ET = E + N;
  if (e >= ET) return;
  int s, d;
  if (e < E) { s = ei[e]; d = ei[E + e]; } else { s = e - E; d = s; }
  const int h = f / D;
  float w = ebuf[e * Hh + h];
  float v = Hm[(size_t)s * F + f] * w;
  atomicAdd(&acc[(size_t)d * F + f], v);
}

// ---------------------------------------------------------------------------
// out = acc / den[dst-head] + bias, optional ReLU (in place)
// ---------------------------------------------------------------------------
__global__ void __launch_bounds__(256)
normalize_bias(float* __restrict__ acc, const float* __restrict__ den,
               const float* __restrict__ bias, int N, int Hh, int D, int do_relu) {
  const int F = Hh * D;
  int idx = blockIdx.x * 256 + threadIdx.x;
  if (idx >= N * F) return;
  int n = idx / F, f = idx - n * F;
  float v = acc[idx] / den[n * Hh + f / D] + bias[f];
  if (do_relu) v = fmaxf(v, 0.0f);
  acc[idx] = v;
}

// ---------------------------------------------------------------------------
// launch
// ---------------------------------------------------------------------------
extern "C" void kernel_launch(void* const* d_in, const int* in_sizes, int n_in,
                              void* d_out, int out_size, void* d_ws, size_t ws_size,
                              hipStream_t stream) {
  const float* x    = (const float*)d_in[0];
  const int*   ei   = (const int*)  d_in[1];
  const float* W1   = (const float*)d_in[2];
  const float* aS1  = (const float*)d_in[3];
  const float* aD1  = (const float*)d_in[4];
  const float* b1   = (const float*)d_in[5];
  const float* W2   = (const float*)d_in[6];
  const float* aS2  = (const float*)d_in[7];
  const float* aD2  = (const float*)d_in[8];
  const float* b2   = (const float*)d_in[9];
  float* out = (float*)d_out;

  const int F1  = in_sizes[5];            // 256 = heads1*D1
  const int CIN = in_sizes[2] / F1;       // 128
  const int N   = in_sizes[0] / CIN;      // 50000
  const int E   = in_sizes[1] / 2;        // 800000
  const int F2  = in_sizes[9];            // 64
  const int H1h = 4;                      // reference: heads=4
  const int D1  = F1 / H1h;               // 64
  const int H2h = 1;
  const int D2  = F2;
  const int ET  = E + N;

  // workspace layout (floats)
  float* ws = (float*)d_ws;
  size_t o = 0;
  float* H1  = ws + o; o += (size_t)N * F1;
  float* X2  = ws + o; o += (size_t)N * F1;   // acc1, then normalized layer-1 output
  float* as1 = ws + o; o += (size_t)N * H1h;
  float* ad1 = ws + o; o += (size_t)N * H1h;
  float* mx1 = ws + o; o += (size_t)N * H1h;
  float* dn1 = ws + o; o += (size_t)N * H1h;
  float* ex1 = ws + o; o += (size_t)ET * H1h;
  float* H2  = ws + o; o += (size_t)N * F2;
  float* as2 = ws + o; o += (size_t)N;
  float* ad2 = ws + o; o += (size_t)N;
  float* mx2 = ws + o; o += (size_t)N;
  float* dn2 = ws + o; o += (size_t)N;
  float* ex2 = ws + o; o += (size_t)ET;

  const float NEG_INF = -__builtin_huge_valf();
  auto blocks = [](int n) { return (n + 255) / 256; };

  // ---- init (every call: atomics accumulate, graph replay must be clean) ----
  fill_kernel<<<blocks(N * H1h), 256, 0, stream>>>(mx1, NEG_INF, N * H1h);
  fill_kernel<<<blocks(N * H1h), 256, 0, stream>>>(dn1, 0.0f, N * H1h);
  fill_kernel<<<blocks(N * F1),  256, 0, stream>>>(X2,  0.0f, N * F1);
  fill_kernel<<<blocks(N * H2h), 256, 0, stream>>>(mx2, NEG_INF, N * H2h);
  fill_kernel<<<blocks(N * H2h), 256, 0, stream>>>(dn2, 0.0f, N * H2h);
  fill_kernel<<<blocks(N * F2),  256, 0, stream>>>(out, 0.0f, N * F2);

  // ---- layer 1 ----
  {
    int strips = (N / 16 + 7) / 8;
    gemm_wmma_f32<<<strips * (F1 / 16), 256, 0, stream>>>(x, W1, H1, N, CIN, F1);
  }
  alpha_kernel<<<blocks(N * H1h), 256, 0, stream>>>(H1, aS1, aD1, as1, ad1, N, H1h, D1);
  edge_logits_max<<<blocks(ET * H1h), 256, 0, stream>>>(ei, E, N, H1h, as1, ad1, ex1, mx1);
  edge_exp_sum<<<blocks(ET * H1h), 256, 0, stream>>>(ei, E, N, H1h, ex1, mx1, dn1);
  {
    int epb = 256 / F1;   // 1
    edge_aggregate<<<(ET + epb - 1) / epb, 256, 0, stream>>>(ei, E, N, H1h, D1, H1, ex1, X2);
  }
  normalize_bias<<<blocks(N * F1), 256, 0, stream>>>(X2, dn1, b1, N, H1h, D1, /*relu=*/1);

  // ---- layer 2 ----
  {
    int strips = (N / 16 + 7) / 8;
    gemm_wmma_f32<<<strips * (F2 / 16), 256, 0, stream>>>(X2, W2, H2, N, F1, F2);
  }
  alpha_kernel<<<blocks(N * H2h), 256, 0, stream>>>(H2, aS2, aD2, as2, ad2, N, H2h, D2);
  edge_logits_max<<<blocks(ET * H2h), 256, 0, stream>>>(ei, E, N, H2h, as2, ad2, ex2, mx2);
  edge_exp_sum<<<blocks(ET * H2h), 256, 0, stream>>>(ei, E, N, H2h, ex2, mx2, dn2);
  {
    int epb = 256 / F2;   // 4
    edge_aggregate<<<(ET + epb - 1) / epb, 256, 0, stream>>>(ei, E, N, H2h, D2, H2, ex2, out);
  }
  normalize_bias<<<blocks(N * F2), 256, 0, stream>>>(out, dn2, b2, N, H2h, D2, /*relu=*/0);
}